// LSTMDecoder_20023137534941
// MI455X (gfx1250) — compile-verified
//
#include <hip/hip_runtime.h>

// ---------------------------------------------------------------------------
// LSTM decoder, fully fused for MI455X (gfx1250, wave32, WMMA).
//   Phase 1: x_gates = [z|last_pos|last_vel] @ W_ih^T + b_ih + b_hh
//   Phase 2: 60-step LSTM recurrence + per-step MLP + cumsum, fused.
// Each workgroup owns 16 batch rows; h/c live in LDS for all 60 steps
// (CDNA5's 320KB-per-WGP LDS). GEMMs use V_WMMA_F32_16X16X4_F32 (exact fp32).
// K trip counts are runtime kernel args (prevents full unroll + LICM spills).
// Fragment streams are hand ping-pong double-buffered: each half issues the
// next 9 loads, then runs 8 WMMAs on the other buffer set (no reg-move chain).
// Phase-1 stages z_aug into LDS with GLOBAL_LOAD_ASYNC_TO_LDS_B128 when the
// toolchain exposes it (ASYNCcnt path), else a float2 copy fallback.
// ---------------------------------------------------------------------------

typedef float v2f __attribute__((ext_vector_type(2)));
typedef float v8f __attribute__((ext_vector_type(8)));
typedef int   v4i __attribute__((ext_vector_type(4)));

// Builtin param types discovered by compile-probe: (v4i AS1*, v4i AS3*, int, int)
typedef __attribute__((address_space(1))) v4i* gptr_b128;
typedef __attribute__((address_space(3))) v4i* lptr_b128;

#define B_   4096
#define H_   1024
#define T_   60
#define G_   4096   // 4*H
#define IN_  1028   // H + 4
#define MH_  64

// LDS row strides (floats): even (8B-aligned float2 frag loads) and with a
// mod-64 residue whose multiples 0..15 are distinct (bank-conflict-free
// across the 16 rows touched by one WMMA fragment load).
#define HS 1026     // 1026 % 64 == 2
#define ZS 1036     // 1036 % 64 == 12; also %4==0 -> 16B-aligned rows (B128)
#define AS 66

#if defined(__has_builtin)
#if __has_builtin(__builtin_amdgcn_global_load_async_to_lds_b128)
#define HAVE_ASYNC_LDS 1
#endif
#if __has_builtin(__builtin_amdgcn_s_wait_asynccnt)
#define HAVE_WAIT_ASYNC 1
#endif
#endif

__device__ __forceinline__ v8f wmma4(v2f a, v2f b, v8f c) {
  // D = A(16x4,f32) * B(4x16,f32) + C(16x16,f32)
  return __builtin_amdgcn_wmma_f32_16x16x4_f32(
      /*neg_a=*/false, a, /*neg_b=*/false, b,
      /*c_mod=*/(short)0, c, /*reuse_a=*/false, /*reuse_b=*/false);
}

__device__ __forceinline__ float sigm(float x) {
  return 1.0f / (1.0f + __expf(-x));
}
__device__ __forceinline__ float tanh_f(float x) {
  float e = __expf(-2.0f * fabsf(x));       // e in (0,1], overflow-safe
  float t = (1.0f - e) / (1.0f + e);
  return copysignf(t, x);
}

// ---------------------------------------------------------------------------
// Phase 1: x_gates[B][4096] = z_aug @ W_ih^T + (b_ih + b_hh)
// Grid: B/16 workgroups of 256 threads (8 waves).
// Each wave: 8 groups of 4 N-tiles (32 accumulator VGPRs / group).
// ---------------------------------------------------------------------------
__global__ __launch_bounds__(256) void xgates_kernel(
    const float* __restrict__ z, const float* __restrict__ last_vel,
    const float* __restrict__ last_pos, const float* __restrict__ W_ih,
    const float* __restrict__ b_ih, const float* __restrict__ b_hh,
    float* __restrict__ xg, int kiters /* = IN_/4 = 257, runtime */) {
  __shared__ float sz[16 * ZS];  // z_aug slab: 16 rows x 1028 (padded)

  const int tid  = threadIdx.x;
  const int wave = tid >> 5;
  const int lane = tid & 31;
  const int lm   = lane & 15;   // fragment row/col index
  const int lh   = lane >> 4;   // fragment K-half
  const int rowbase = blockIdx.x * 16;

  // ---- stage z_aug slab into LDS ----------------------------------------
#if defined(HAVE_ASYNC_LDS)
  // 16 rows x 1024 floats as 4096 B128 chunks, direct global->LDS (ASYNCcnt).
  for (int i = tid; i < 16 * 256; i += 256) {
    int r = i >> 8, c4 = i & 255;
    __builtin_amdgcn_global_load_async_to_lds_b128(
        (gptr_b128)(z + (size_t)(rowbase + r) * H_ + c4 * 4),
        (lptr_b128)(&sz[r * ZS + c4 * 4]),
        0, 0);
  }
#else
  for (int i = tid; i < 16 * 512; i += 256) {
    int r = i >> 9, c2 = i & 511;
    *(v2f*)(&sz[r * ZS + c2 * 2]) =
        *(const v2f*)(z + (size_t)(rowbase + r) * H_ + c2 * 2);
  }
#endif
  if (tid < 32) {
    int r = tid & 15, p = tid >> 4;
    const float* src = p ? last_vel : last_pos;  // cols 1024-25 pos, 1026-27 vel
    sz[r * ZS + 1024 + p * 2 + 0] = src[(size_t)(rowbase + r) * 2 + 0];
    sz[r * ZS + 1024 + p * 2 + 1] = src[(size_t)(rowbase + r) * 2 + 1];
  }
#if defined(HAVE_ASYNC_LDS)
#if defined(HAVE_WAIT_ASYNC)
  __builtin_amdgcn_s_wait_asynccnt(0);
#else
  asm volatile("s_wait_asynccnt 0" ::: "memory");
#endif
#endif
  __syncthreads();

  const float* ap = sz + lm * ZS + lh * 2;

#pragma unroll 1
  for (int grp = 0; grp < 8; ++grp) {
    asm volatile("" ::: "memory");   // keep LDS frag loads inside this group
    const int nt0 = wave * 32 + grp * 4;       // first N-tile of group

    v8f acc[4];
#pragma unroll
    for (int j = 0; j < 4; ++j) {
      int g0 = (nt0 + j) * 16 + lm;
      float bias = b_ih[g0] + b_hh[g0];
#pragma unroll
      for (int r = 0; r < 8; ++r) acc[j][r] = bias;
    }

    // 4 B streams, constant offsets of j*16 rows (j*16*IN_ floats)
    const float* bbase = W_ih + (size_t)(nt0 * 16 + lm) * IN_ + lh * 2;

    auto loadB = [&](v2f(&buf)[4], int k) {
#pragma unroll
      for (int j = 0; j < 4; ++j)
        buf[j] = *(const v2f*)(bbase + (size_t)j * 16 * IN_ + (size_t)k * 4);
    };
    auto runWmma = [&](v2f a, v2f(&buf)[4]) {
#pragma unroll
      for (int j = 0; j < 4; ++j) acc[j] = wmma4(a, buf[j], acc[j]);
    };

    v2f a0 = *(const v2f*)ap, a1;
    v2f b0[4], b1[4];
    loadB(b0, 0);
    int k = 1;
#pragma unroll 1
    for (; k + 1 < kiters; k += 2) {
      a1 = *(const v2f*)(ap + (size_t)k * 4);
      loadB(b1, k);
      runWmma(a0, b0);
      a0 = *(const v2f*)(ap + (size_t)(k + 1) * 4);
      loadB(b0, k + 1);
      runWmma(a1, b1);
    }
#pragma unroll 1
    for (; k < kiters; ++k) {        // remainder (kiters odd: 0 iterations)
      a1 = *(const v2f*)(ap + (size_t)k * 4);
      loadB(b1, k);
      runWmma(a0, b0);
      a0 = a1;
#pragma unroll
      for (int j = 0; j < 4; ++j) b0[j] = b1[j];
    }
    runWmma(a0, b0);

#pragma unroll
    for (int j = 0; j < 4; ++j)
#pragma unroll
      for (int r = 0; r < 8; ++r)
        xg[(size_t)(rowbase + r + lh * 8) * G_ + (nt0 + j) * 16 + lm] =
            acc[j][r];
  }
}

// ---------------------------------------------------------------------------
// Phase 2: fused 60-step LSTM + MLP + cumsum.
// Grid: B/16 workgroups of 256 threads. h_cur/h_nxt/c in LDS (~197KB).
// Per step each wave runs 4 passes; a pass owns 2 adjacent 16-wide h-column
// chunks x 4 gates = 8 accumulator tiles (64 VGPRs). One LDS A-fragment
// feeds 8 WMMAs; 8 independent L2-resident B streams are ping-pong buffered.
// ---------------------------------------------------------------------------
__global__ __launch_bounds__(256) void lstm_fused(
    const float* __restrict__ xg, const float* __restrict__ W_hh,
    const float* __restrict__ W1, const float* __restrict__ b1,
    const float* __restrict__ W2, const float* __restrict__ b2,
    const float* __restrict__ last_pos, float* __restrict__ out,
    int kiters /* = H_/4 = 256, runtime */) {
  __shared__ float sh[2][16 * HS];   // h double buffer
  __shared__ float sc[16 * HS];      // c
  __shared__ float sact[16 * AS];    // relu(h@W1^T+b1), 16x64 padded
  __shared__ float scum[32];         // running cumsum (+last_pos), [o*16+b]

  const int tid  = threadIdx.x;
  const int wave = tid >> 5;
  const int lane = tid & 31;
  const int lm   = lane & 15;
  const int lh   = lane >> 4;
  const int rowbase = blockIdx.x * 16;

  for (int i = tid; i < 16 * HS; i += 256) { sh[0][i] = 0.0f; sc[i] = 0.0f; }
  if (tid < 32)
    scum[tid] = last_pos[(size_t)(rowbase + lm) * 2 + lh];
  __syncthreads();

  const float* xgp = xg + (size_t)rowbase * G_;
  int cur = 0;

  for (int t = 0; t < T_; ++t) {
    const float* h_cur = &sh[cur][0];
    float*       h_nxt = &sh[cur ^ 1][0];
    const float* ap = h_cur + lm * HS + lh * 2;

    // ---- gates = x_gates + h_cur @ W_hh^T, then cell update --------------
#pragma unroll 1
    for (int g = 0; g < 4; ++g) {
      asm volatile("" ::: "memory");  // keep LDS frag loads inside this pass
      const int jcol0 = (wave * 8 + g * 2) * 16;  // first of 2 chunks

      v8f acc[8];  // [gate*2 + chunk]
#pragma unroll
      for (int gI = 0; gI < 4; ++gI)
#pragma unroll
        for (int cj = 0; cj < 2; ++cj) {
          const int col = gI * H_ + jcol0 + cj * 16 + lm;
#pragma unroll
          for (int r = 0; r < 8; ++r)
            acc[gI * 2 + cj][r] = xgp[(size_t)(r + lh * 8) * G_ + col];
        }

      // 8 B streams off one base: +gate*4MB +chunk*64KB (const offsets)
      const float* bbase = W_hh + (size_t)(jcol0 + lm) * H_ + lh * 2;

      auto loadB = [&](v2f(&buf)[8], int k) {
#pragma unroll
        for (int gI = 0; gI < 4; ++gI)
#pragma unroll
          for (int cj = 0; cj < 2; ++cj)
            buf[gI * 2 + cj] = *(const v2f*)(bbase + (size_t)gI * H_ * H_ +
                                             cj * 16 * H_ + (size_t)k * 4);
      };
      auto runWmma = [&](v2f a, v2f(&buf)[8]) {
#pragma unroll
        for (int i = 0; i < 8; ++i) acc[i] = wmma4(a, buf[i], acc[i]);
      };

      v2f a0 = *(const v2f*)ap, a1;
      v2f b0[8], b1[8];
      loadB(b0, 0);
      int k = 1;
#pragma unroll 1
      for (; k + 1 < kiters; k += 2) {
        a1 = *(const v2f*)(ap + (size_t)k * 4);
        loadB(b1, k);
        runWmma(a0, b0);
        a0 = *(const v2f*)(ap + (size_t)(k + 1) * 4);
        loadB(b0, k + 1);
        runWmma(a1, b1);
      }
#pragma unroll 1
      for (; k < kiters; ++k) {      // remainder (kiters even: 1 iteration)
        a1 = *(const v2f*)(ap + (size_t)k * 4);
        loadB(b1, k);
        runWmma(a0, b0);
        a0 = a1;
#pragma unroll
        for (int i = 0; i < 8; ++i) b0[i] = b1[i];
      }
      runWmma(a0, b0);

      // ---- elementwise LSTM cell update for both chunks ------------------
#pragma unroll
      for (int cj = 0; cj < 2; ++cj)
#pragma unroll
        for (int r = 0; r < 8; ++r) {
          const int cidx = (r + lh * 8) * HS + jcol0 + cj * 16 + lm;
          float iv = sigm(acc[0 * 2 + cj][r]);
          float fv = sigm(acc[1 * 2 + cj][r]);
          float gv = tanh_f(acc[2 * 2 + cj][r]);
          float ov = sigm(acc[3 * 2 + cj][r]);
          float cv = fv * sc[cidx] + iv * gv;
          sc[cidx] = cv;
          h_nxt[cidx] = ov * tanh_f(cv);
        }
    }
    __syncthreads();

    // ---- MLP layer 1: act = relu(h_nxt @ W1^T + b1), waves 0-3 -----------
    if (wave < 4) {
      asm volatile("" ::: "memory");
      const int nb = wave * 16 + lm;               // output col 0..63
      v8f a1v;
      const float bias = b1[nb];
#pragma unroll
      for (int r = 0; r < 8; ++r) a1v[r] = bias;
      const float* hp  = h_nxt + lm * HS + lh * 2;
      const float* w1p = W1 + (size_t)nb * H_ + lh * 2;

      v2f ac = *(const v2f*)(hp);
      v2f bc = *(const v2f*)(w1p);
#pragma unroll 2
      for (int k = 1; k < kiters; ++k) {
        v2f an = *(const v2f*)(hp + (size_t)k * 4);
        v2f bn = *(const v2f*)(w1p + (size_t)k * 4);
        a1v = wmma4(ac, bc, a1v);
        ac = an;
        bc = bn;
      }
      a1v = wmma4(ac, bc, a1v);
#pragma unroll
      for (int r = 0; r < 8; ++r)
        sact[(r + lh * 8) * AS + nb] = fmaxf(a1v[r], 0.0f);
    }
    __syncthreads();

    // ---- MLP layer 2 + cumsum: wave 0 ------------------------------------
    if (tid < 32) {
      const int b = lm, o = lh;
      float d = b2[o];
      const float* w2p = W2 + o * MH_;
#pragma unroll 8
      for (int m = 0; m < MH_; ++m) d = fmaf(sact[b * AS + m], w2p[m], d);
      float cumv = scum[tid] + d;
      scum[tid] = cumv;
      out[((size_t)(rowbase + b) * T_ + t) * 2 + o] = cumv;
    }
    __syncthreads();

    cur ^= 1;
  }
}

// ---------------------------------------------------------------------------
extern "C" void kernel_launch(void* const* d_in, const int* in_sizes, int n_in,
                              void* d_out, int out_size, void* d_ws, size_t ws_size,
                              hipStream_t stream) {
  const float* z   = (const float*)d_in[0];
  const float* lv  = (const float*)d_in[1];   // last_vel
  const float* lp  = (const float*)d_in[2];   // last_pos
  const float* Wih = (const float*)d_in[3];
  const float* Whh = (const float*)d_in[4];
  const float* bih = (const float*)d_in[5];
  const float* bhh = (const float*)d_in[6];
  const float* W1  = (const float*)d_in[7];
  const float* b1  = (const float*)d_in[8];
  const float* W2  = (const float*)d_in[9];
  const float* b2  = (const float*)d_in[10];
  float* out = (float*)d_out;
  float* xg  = (float*)d_ws;                  // 4096*4096 fp32 = 64 MB scratch

  dim3 blk(256);
  dim3 grd(B_ / 16);
  xgates_kernel<<<grd, blk, 0, stream>>>(z, lv, lp, Wih, bih, bhh, xg,
                                         IN_ / 4);
  lstm_fused<<<grd, blk, 0, stream>>>(xg, Whh, W1, b1, W2, b2, lp, out,
                                      H_ / 4);
}